// RoiPooling_32384053412535
// MI455X (gfx1250) — compile-verified
//
#include <hip/hip_runtime.h>
#include <hip/hip_bf16.h>

typedef float v2f __attribute__((ext_vector_type(2)));
typedef float v4f __attribute__((ext_vector_type(4)));
typedef float v8f __attribute__((ext_vector_type(8)));

#define HW      128
#define CCH     256
#define CHUNK   128      // channels per LDS pass
#define TSTRIDE 144      // padded LDS row stride in floats (bank-conflict free B reads)
#define NROWS   112      // 16 padded k-slots * 7 ax columns

// Mirror of the reference axis_samples() in f32.
__device__ __forceinline__ void axis_samples(int lo, int hi, int a,
                                             int& i0g, int& i1g, float& frac) {
    float span = (float)(hi - lo);
    float src  = ((float)a + 0.5f) * (span / 7.0f) - 0.5f;
    float hic  = span - 1.0f; if (hic < 0.0f) hic = 0.0f;
    src = fminf(fmaxf(src, 0.0f), hic);
    int i0 = (int)floorf(src);
    int i1 = i0 + 1;
    int lim = (hi - lo) - 1;
    if (i1 > lim) i1 = lim;
    frac = src - (float)i0;
    i0g = i0 + lo;
    i1g = i1 + lo;
}

__global__ __launch_bounds__(256)
void RoiAlignWmmaKernel(const float* __restrict__ fm,
                        const int*   __restrict__ boxes,
                        float*       __restrict__ out,
                        int nboxes) {
    __shared__ float t[NROWS * TSTRIDE];   // 63 KB: x-blended rows, padded K=16

    const int b = blockIdx.x;
    if (b >= nboxes) return;

    const int tid  = threadIdx.x;      // 0..255 (8 wave32s)
    const int lane = tid & 31;
    const int wave = tid >> 5;

    const int x1 = boxes[b * 4 + 0];
    const int y1 = boxes[b * 4 + 1];
    const int x2 = boxes[b * 4 + 2];
    const int y2 = boxes[b * 4 + 3];

    // ---- Per-lane A-matrix fragments (16x4 f32 layout, ISA 7.12.2) ----
    // Wy[m,k] = (k==m)?(1-fy[m]) : (k==m+7)?fy[m] : 0   (m<7, k<14; else 0)
    const int mrow = lane & 15;
    const int khi  = lane >> 4;        // 0: K={0,1}(+4q), 1: K={2,3}(+4q)
    int ys0_m, ys1_m; float fy_m;
    axis_samples(y1, y2, (mrow < 7 ? mrow : 0), ys0_m, ys1_m, fy_m);
    v2f apre[4];
    for (int q = 0; q < 4; ++q) {
        int k0 = 4 * q + 2 * khi;
        float a0 = 0.0f, a1 = 0.0f;
        if (mrow < 7) {
            if (k0     == mrow)     a0 = 1.0f - fy_m;
            else if (k0     == mrow + 7) a0 = fy_m;
            if (k0 + 1 == mrow)     a1 = 1.0f - fy_m;
            else if (k0 + 1 == mrow + 7) a1 = fy_m;
        }
        apre[q].x = a0; apre[q].y = a1;
    }

    // Pass-1 mapping: 32 lanes * float4 cover one 128-channel chunk; 8 row-groups
    const int c4   = (tid & 31) * 4;
    const int rgrp = tid >> 5;

    for (int chunk = 0; chunk < 2; ++chunk) {
        const int cbase = chunk * CHUNK;

        // ---------- Pass 1: x-blend the <=14 source rows into LDS ----------
        for (int i = 0; i < 14; ++i) {
            int r  = rgrp * 14 + i;        // 0..111
            int k  = r / 7;                // padded K slot 0..15
            int ax = r - k * 7;
            v4f val = {};
            if (k < 14) {
                int ky = (k < 7) ? k : (k - 7);
                int ys0, ys1; float fy;
                axis_samples(y1, y2, ky, ys0, ys1, fy);
                int j = (k < 7) ? ys0 : ys1;
                int xs0, xs1; float fx;
                axis_samples(x1, x2, ax, xs0, xs1, fx);
                const v4f v0 = *(const v4f*)(fm + (size_t)(j * HW + xs0) * CCH + cbase + c4);
                const v4f v1 = *(const v4f*)(fm + (size_t)(j * HW + xs1) * CCH + cbase + c4);
                val = v0 * (1.0f - fx) + v1 * fx;
            }
            *(v4f*)&t[r * TSTRIDE + c4] = val;
        }
        __syncthreads();

        // ---------- Pass 2: y-blend = (16x16) <- (16x4)x(4x16) WMMA, 4 K-steps ----------
        for (int j2 = 0; j2 < 7; ++j2) {
            const int tau = wave * 7 + j2;     // 0..55 N-tiles (uniform per wave)
            const int ax  = tau >> 3;
            const int c0  = (tau & 7) * 16;
            const int n   = lane & 15;

            v8f acc = {};
#if __has_builtin(__builtin_amdgcn_wmma_f32_16x16x4_f32)
            for (int q = 0; q < 4; ++q) {
                int kk = 4 * q + 2 * khi;
                v2f bb;
                bb.x = t[(kk       * 7 + ax) * TSTRIDE + c0 + n];
                bb.y = t[((kk + 1) * 7 + ax) * TSTRIDE + c0 + n];
                acc = __builtin_amdgcn_wmma_f32_16x16x4_f32(
                        false, apre[q], false, bb, (short)0, acc, false, false);
            }
            if (lane < 16) {
                for (int r = 0; r < 7; ++r) {
                    out[((size_t)b * 49 + r * 7 + ax) * CCH + cbase + c0 + lane] = acc[r];
                }
            }
#else
            // VALU fallback (same math) — keeps file compiling if builtin is absent
            if (lane < 16) {
                for (int r = 0; r < 7; ++r) {
                    int ys0, ys1; float fy;
                    axis_samples(y1, y2, r, ys0, ys1, fy);
                    float top = t[(r       * 7 + ax) * TSTRIDE + c0 + lane];
                    float bot = t[((r + 7) * 7 + ax) * TSTRIDE + c0 + lane];
                    out[((size_t)b * 49 + r * 7 + ax) * CCH + cbase + c0 + lane] =
                        top * (1.0f - fy) + bot * fy;
                }
            }
            (void)acc;
#endif
        }
        __syncthreads();
    }
}

extern "C" void kernel_launch(void* const* d_in, const int* in_sizes, int n_in,
                              void* d_out, int out_size, void* d_ws, size_t ws_size,
                              hipStream_t stream) {
    const float* fm    = (const float*)d_in[0];   // (1,128,128,256) f32
    const int*   boxes = (const int*)d_in[1];     // (N,4) i32
    // d_in[2] = anchor_size == 7 (baked into kernel constants)
    int nboxes = in_sizes[1] / 4;
    float* out = (float*)d_out;                   // (N,1,7,7,256) f32

    RoiAlignWmmaKernel<<<dim3(nboxes), dim3(256), 0, stream>>>(fm, boxes, out, nboxes);
}